// GCN_704374637025
// MI455X (gfx1250) — compile-verified
//
#include <hip/hip_runtime.h>
#include <hip/hip_bf16.h>

typedef __attribute__((ext_vector_type(2))) float v2f;
typedef __attribute__((ext_vector_type(8))) float v8f;

#define FIN 128

// ---------------------------------------------------------------------------
// WMMA GEMM: C[nrows x FOUT] = act(A[nrows x 128]) @ W[128 x FOUT]
// One wave -> 16-row stripe, all FOUT columns (FOUT/16 accumulator tiles).
// W staged in LDS. fp32 WMMA 16x16x4 per ISA layout:
//   A frag: VGPR j, lane l: A[M = l%16][K = k + j + 2*(l>=16)]
//   B frag: VGPR j, lane l: W[K = k + j + 2*(l>=16)][N = t*16 + l%16]
//   C/D   : VGPR r, lane l: C[M = r + 8*(l>=16)][N = t*16 + l%16]
// ---------------------------------------------------------------------------
template <int FOUT, bool RELU>
__global__ __launch_bounds__(256) void gemm_wmma_kernel(
    const float* __restrict__ A, const float* __restrict__ W,
    float* __restrict__ C, int nrows)
{
    constexpr int NT = FOUT / 16;
    __shared__ float ldsW[FIN * FOUT];   // 64KB (FOUT=128) / 32KB (FOUT=64)

    // cooperative load of W into LDS (row-major)
    for (int i = threadIdx.x; i < FIN * FOUT; i += 256)
        ldsW[i] = W[i];
    __syncthreads();

    const int wave = threadIdx.x >> 5;
    const int lane = threadIdx.x & 31;
    const int half = lane >> 4;          // 0: lanes 0-15, 1: lanes 16-31
    const int m    = lane & 15;

    const int tile   = blockIdx.x * 8 + wave;
    const int ntiles = (nrows + 15) >> 4;
    if (tile >= ntiles) return;

    const int row0 = tile * 16;
    int arow = row0 + m;
    if (arow >= nrows) arow = nrows - 1;           // clamp pad rows (stores guarded)
    const float* aptr = A + (size_t)arow * FIN;

    v8f acc[NT] = {};

    for (int k = 0; k < FIN; k += 4) {
        const int kk = k + 2 * half;
        float2 af = *(const float2*)(aptr + kk);   // K = kk, kk+1
        if (RELU) {
            af.x = af.x > 0.0f ? af.x : 0.0f;
            af.y = af.y > 0.0f ? af.y : 0.0f;
        }
        v2f a;
        a.x = af.x;
        a.y = af.y;
#pragma unroll
        for (int t = 0; t < NT; ++t) {
            v2f b;
            b.x = ldsW[(kk)     * FOUT + t * 16 + m];
            b.y = ldsW[(kk + 1) * FOUT + t * 16 + m];
            acc[t] = __builtin_amdgcn_wmma_f32_16x16x4_f32(
                false, a, false, b, (short)0, acc[t], false, false);
        }
    }

#pragma unroll
    for (int r = 0; r < 8; ++r) {
        const int row = row0 + r + 8 * half;
        if (row < nrows) {
#pragma unroll
            for (int t = 0; t < NT; ++t)
                C[(size_t)row * FOUT + t * 16 + m] = acc[t][r];
        }
    }
}

// ---------------------------------------------------------------------------
// out[i] = bias[i % FOUT]  (broadcast-init so the scatter adds on top of bias)
// ---------------------------------------------------------------------------
template <int FOUT>
__global__ void fill_bias_kernel(float* __restrict__ out,
                                 const float* __restrict__ b, int total)
{
    const int i = blockIdx.x * blockDim.x + threadIdx.x;
    if (i < total) out[i] = b[i & (FOUT - 1)];
}

// ---------------------------------------------------------------------------
// One wave per edge: out[dst] += h[src] * w   (float atomics, L2-resident)
// ---------------------------------------------------------------------------
template <int F>
__global__ __launch_bounds__(256) void scatter_kernel(
    const float* __restrict__ h, const int* __restrict__ srcI,
    const int* __restrict__ dstI, const float* __restrict__ ew,
    float* __restrict__ out, int nedges)
{
    const int e = blockIdx.x * 8 + (threadIdx.x >> 5);
    if (e >= nedges) return;
    const int   lane = threadIdx.x & 31;
    const int   s = srcI[e];
    const int   d = dstI[e];
    const float w = ew[e];

    constexpr int PER = F / 32;                    // 4 (F=128) or 2 (F=64)
    const float* hp = h   + (size_t)s * F + lane * PER;
    float*       op = out + (size_t)d * F + lane * PER;

    if constexpr (PER == 4) {
        float4 v = *(const float4*)hp;
        unsafeAtomicAdd(op + 0, v.x * w);
        unsafeAtomicAdd(op + 1, v.y * w);
        unsafeAtomicAdd(op + 2, v.z * w);
        unsafeAtomicAdd(op + 3, v.w * w);
    } else {
        float2 v = *(const float2*)hp;
        unsafeAtomicAdd(op + 0, v.x * w);
        unsafeAtomicAdd(op + 1, v.y * w);
    }
}

// ---------------------------------------------------------------------------
extern "C" void kernel_launch(void* const* d_in, const int* in_sizes, int n_in,
                              void* d_out, int out_size, void* d_ws, size_t ws_size,
                              hipStream_t stream)
{
    const float* x  = (const float*)d_in[0];
    const int*   ei = (const int*)d_in[1];
    const float* ew = (const float*)d_in[2];
    const float* W1 = (const float*)d_in[3];
    const float* b1 = (const float*)d_in[4];
    const float* W2 = (const float*)d_in[5];
    const float* b2 = (const float*)d_in[6];
    const float* W3 = (const float*)d_in[7];
    const float* b3 = (const float*)d_in[8];
    float* out = (float*)d_out;

    const int N = in_sizes[0] / FIN;
    const int E = in_sizes[2];
    const int* srcI = ei;         // edge_index row 0
    const int* dstI = ei + E;     // edge_index row 1

    float* bufA = (float*)d_ws;              // N x 128
    float* bufB = bufA + (size_t)N * FIN;    // N x 128

    const int ntiles     = (N + 15) / 16;
    const int gemmBlocks = (ntiles + 7) / 8;
    const int edgeBlocks = (E + 7) / 8;

    // ---- layer 1: h1 = x @ W1 ; B = b1 + scatter(h1) -----------------------
    gemm_wmma_kernel<128, false><<<gemmBlocks, 256, 0, stream>>>(x, W1, bufA, N);
    fill_bias_kernel<128><<<(N * 128 + 255) / 256, 256, 0, stream>>>(bufB, b1, N * 128);
    scatter_kernel<128><<<edgeBlocks, 256, 0, stream>>>(bufA, srcI, dstI, ew, bufB, E);

    // ---- layer 2: h2 = relu(B) @ W2 ; B = b2 + scatter(h2) -----------------
    gemm_wmma_kernel<128, true><<<gemmBlocks, 256, 0, stream>>>(bufB, W2, bufA, N);
    fill_bias_kernel<128><<<(N * 128 + 255) / 256, 256, 0, stream>>>(bufB, b2, N * 128);
    scatter_kernel<128><<<edgeBlocks, 256, 0, stream>>>(bufA, srcI, dstI, ew, bufB, E);

    // ---- layer 3: h3 = relu(B) @ W3 ; out = b3 + scatter(h3) ---------------
    gemm_wmma_kernel<64, true><<<gemmBlocks, 256, 0, stream>>>(bufB, W3, bufA, N);
    fill_bias_kernel<64><<<(N * 64 + 255) / 256, 256, 0, stream>>>(out, b3, N * 64);
    scatter_kernel<64><<<edgeBlocks, 256, 0, stream>>>(bufA, srcI, dstI, ew, out, E);
}